// QuantizerBottleneck_86569360818548
// MI455X (gfx1250) — compile-verified
//
#include <hip/hip_runtime.h>

typedef __attribute__((ext_vector_type(2))) float v2f;
typedef __attribute__((ext_vector_type(8))) float v8f;

#define B_  16
#define D_  64
#define T_  2048
#define NQ_ 8
#define K_  2048
#define M_  (B_ * T_)            // 32768 rows

#define CHUNK_TILES  8                       // N-tiles staged per buffer
#define CHUNK_CODES  (CHUNK_TILES * 16)      // 128 codes
#define CHUNK_BYTES  (CHUNK_CODES * D_ * 4)  // 32 KB source bytes
#define ROW_PAD_F    68                      // 64 floats + 4 pad -> conflict-free banks
#define BUF_FLOATS   (CHUNK_CODES * ROW_PAD_F)
#define NCHUNKS      (K_ / CHUNK_CODES)      // 16

// ---------------------------------------------------------------------------
// [B,D,T] -> [B*T, D] row-major residual scratch
// ---------------------------------------------------------------------------
__global__ void rvq_transpose(const float* __restrict__ x, float* __restrict__ res) {
    int idx = blockIdx.x * blockDim.x + threadIdx.x;   // over B*D*T, t fastest
    if (idx >= B_ * D_ * T_) return;
    int t = idx % T_;
    int d = (idx / T_) % D_;
    int b = idx / (T_ * D_);
    res[(size_t)(b * T_ + t) * D_ + d] = x[idx];
}

// ---------------------------------------------------------------------------
// ||cb||^2 for every code of every level (NQ*K values)
// ---------------------------------------------------------------------------
__global__ void rvq_norms(const float* __restrict__ cbs, float* __restrict__ norms) {
    int idx = blockIdx.x * blockDim.x + threadIdx.x;   // over NQ*K
    if (idx >= NQ_ * K_) return;
    const float4* row = (const float4*)(cbs + (size_t)idx * D_);
    float s = 0.f;
#pragma unroll
    for (int i = 0; i < D_ / 4; ++i) {
        float4 v = row[i];
        s += v.x * v.x + v.y * v.y + v.z * v.z + v.w * v.w;
    }
    norms[idx] = s;
}

// Async-copy one 128-code chunk into a padded LDS buffer (all 256 threads).
// Each 16B unit u of the source goes to LDS row (u/16) * 272B + (u%16)*16B.
__device__ __forceinline__ void stage_chunk(const float* __restrict__ src,
                                            unsigned lds_base) {
#pragma unroll
    for (int j = 0; j < (CHUNK_BYTES / 16) / 256; ++j) {   // 8 ops/thread
        unsigned u   = j * 256 + threadIdx.x;              // 16-byte unit id
        unsigned row = u >> 4;
        unsigned in  = u & 15;
        unsigned la  = lds_base + row * (ROW_PAD_F * 4) + in * 16;
        unsigned long long ga = (unsigned long long)(const char*)src + u * 16ull;
        asm volatile("global_load_async_to_lds_b128 %0, %1, off"
                     :: "v"(la), "v"(ga) : "memory");
    }
}

// ---------------------------------------------------------------------------
// One quantization level: per 16-row tile, WMMA distance GEMM over all K=2048
// codes (codebook staged in LDS, double-buffered async prefetch) + fused
// running argmin + in-place residual update. One wave per M-tile, 8/block.
// ---------------------------------------------------------------------------
__global__ __launch_bounds__(256) void rvq_level(float* __restrict__ res,
                                                 const float* __restrict__ cb,
                                                 const float* __restrict__ cnorm) {
    __shared__ float sbuf[2][BUF_FLOATS];    // ~68 KB total

    const int lane  = threadIdx.x & 31;
    const int wave  = threadIdx.x >> 5;
    const int l16   = lane & 15;
    const int half  = lane >> 4;             // 0: lanes 0-15, 1: lanes 16-31
    const int mtile = blockIdx.x * 8 + wave; // 0..2047

    // A tile (16 rows x 64 dims), ISA 32-bit A layout for 16x16x4:
    // lane l16 / l16+16 holds row l16, dims {4k+2*half, 4k+1+2*half}
    const int rowA = mtile * 16 + l16;
    const float* aptr = res + (size_t)rowA * D_ + 2 * half;
    v2f a[16];
#pragma unroll
    for (int k = 0; k < 16; ++k) a[k] = *(const v2f*)(aptr + 4 * k);

    float best_d[8];
    int   best_n[8];
#pragma unroll
    for (int r = 0; r < 8; ++r) { best_d[r] = 3.4e38f; best_n[r] = 0; }

    // Prefetch chunk 0 into buffer 0
    stage_chunk(cb, (unsigned)(unsigned long long)(uintptr_t)&sbuf[0][0]);

    for (int c = 0; c < NCHUNKS; ++c) {
        asm volatile("s_wait_asynccnt 0x0" ::: "memory");  // my copies done
        __syncthreads();                                   // everyone's done

        if (c + 1 < NCHUNKS)
            stage_chunk(cb + (size_t)(c + 1) * CHUNK_CODES * D_,
                        (unsigned)(unsigned long long)(uintptr_t)&sbuf[(c + 1) & 1][0]);

        const float* buf = &sbuf[c & 1][0];
        for (int ntl = 0; ntl < CHUNK_TILES; ++ntl) {
            const int code = (c * CHUNK_TILES + ntl) * 16 + l16;
            // B layout mirrors A; padded row stride keeps banks conflict-free
            const float* bptr = buf + (ntl * 16 + l16) * ROW_PAD_F + 2 * half;
            v2f bv[16];
#pragma unroll
            for (int k = 0; k < 16; ++k) bv[k] = *(const v2f*)(bptr + 4 * k);

            v8f cacc = {};
#pragma unroll
            for (int k = 0; k < 16; ++k)
                cacc = __builtin_amdgcn_wmma_f32_16x16x4_f32(
                           false, a[k], false, bv[k], (short)0, cacc, false, false);

            // C layout: reg r -> row (r + 8*half), col = tile*16 + l16
            const float cn = cnorm[code];
#pragma unroll
            for (int r = 0; r < 8; ++r) {
                float dist = cn - 2.0f * cacc[r];
                bool upd   = dist < best_d[r];       // strict < keeps lowest index
                best_d[r]  = upd ? dist : best_d[r];
                best_n[r]  = upd ? code : best_n[r];
            }
        }
        __syncthreads();   // all reads of buf[c&1] done before it is refilled
    }

    // Argmin across the 16 lanes of each half (xor masks stay inside the half)
#pragma unroll
    for (int m = 1; m < 16; m <<= 1) {
#pragma unroll
        for (int r = 0; r < 8; ++r) {
            float od = __shfl_xor(best_d[r], m, 32);
            int   on = __shfl_xor(best_n[r], m, 32);
            bool upd = (od < best_d[r]) || (od == best_d[r] && on < best_n[r]);
            best_d[r] = upd ? od : best_d[r];
            best_n[r] = upd ? on : best_n[r];
        }
    }

    // residual[row] -= cb[best]; half h owns rows r+8*h, lane l16 owns 4 dims
#pragma unroll
    for (int r = 0; r < 8; ++r) {
        int row = mtile * 16 + r + 8 * half;
        int idx = best_n[r];                         // uniform within the half
        float4*       rp = (float4*)(res + (size_t)row * D_ + 4 * l16);
        const float4* cp = (const float4*)(cb + (size_t)idx * D_ + 4 * l16);
        float4 rv = *rp, cv = *cp;
        rv.x -= cv.x; rv.y -= cv.y; rv.z -= cv.z; rv.w -= cv.w;
        *rp = rv;
    }
}

// ---------------------------------------------------------------------------
// out[b,d,t] = x[b,d,t] - residual_final[b*T+t, d]   (== sum of codes)
// ---------------------------------------------------------------------------
__global__ void rvq_output(const float* __restrict__ x, const float* __restrict__ res,
                           float* __restrict__ out) {
    int idx = blockIdx.x * blockDim.x + threadIdx.x;
    if (idx >= B_ * D_ * T_) return;
    int t = idx % T_;
    int d = (idx / T_) % D_;
    int b = idx / (T_ * D_);
    out[idx] = x[idx] - res[(size_t)(b * T_ + t) * D_ + d];
}

extern "C" void kernel_launch(void* const* d_in, const int* in_sizes, int n_in,
                              void* d_out, int out_size, void* d_ws, size_t ws_size,
                              hipStream_t stream) {
    const float* x   = (const float*)d_in[0];   // [B, D, T]
    const float* cbs = (const float*)d_in[1];   // [NQ, K, D]
    float* out = (float*)d_out;

    float* res   = (float*)d_ws;                                            // 8 MB
    float* norms = (float*)((char*)d_ws + (size_t)M_ * D_ * sizeof(float)); // 64 KB

    const int n = B_ * D_ * T_;
    rvq_transpose<<<(n + 255) / 256, 256, 0, stream>>>(x, res);
    rvq_norms<<<(NQ_ * K_ + 255) / 256, 256, 0, stream>>>(cbs, norms);

    for (int lv = 0; lv < NQ_; ++lv) {
        rvq_level<<<(M_ / 16) / 8, 256, 0, stream>>>(
            res, cbs + (size_t)lv * K_ * D_, norms + (size_t)lv * K_);
    }

    rvq_output<<<(n + 255) / 256, 256, 0, stream>>>(x, res, out);
}